// GCN1_11321533792937
// MI455X (gfx1250) — compile-verified
//
#include <hip/hip_runtime.h>

typedef __attribute__((ext_vector_type(2))) float v2f;
typedef __attribute__((ext_vector_type(8))) float v8f;

#define N_NODES 40000
#define DIM 128

// ---------------------------------------------------------------------------
// Dense GEMM via V_WMMA_F32_16X16X4_F32:  out[N, NCOLS] = A[N,128] @ W[128,NCOLS] (+bias)
// One wave per 16x16 output tile; blockIdx.x = row tile, wave-in-block = col tile.
// NCOLS is a compile-time constant so all W loads / out stores use immediate
// offsets from a single base pointer (no 64-bit VALU address chains).
// ---------------------------------------------------------------------------
template <int NCOLS, bool HAS_BIAS>
__global__ void gemm_wmma_f32_kernel(const float* __restrict__ A,
                                     const float* __restrict__ W,
                                     const float* __restrict__ bias,
                                     float* __restrict__ out)
{
    const int lane = threadIdx.x & 31;
    const int wv   = threadIdx.x >> 5;      // column tile within block
    const int rt   = blockIdx.x;            // row tile
    const int half = lane >> 4;             // 0: K pair {k,k+1}, 1: K pair {k+2,k+3}
    const int lm   = lane & 15;

    const int colN = wv * 16 + lm;          // B/C/D column for this lane

    // Base pointers: all loop accesses are immediate offsets from these.
    const float* __restrict__ arow = A + (size_t)(rt * 16 + lm) * DIM + 2 * half;
    const float* __restrict__ wcol = W + (size_t)(2 * half) * NCOLS + colN;

    v8f c = {0.f, 0.f, 0.f, 0.f, 0.f, 0.f, 0.f, 0.f};

#pragma unroll
    for (int k = 0; k < DIM; k += 4) {
        v2f a, b;
        a.x = arow[k];                       // A[row][k + 2*half]
        a.y = arow[k + 1];                   // contiguous pair -> global_load_b64
        b.x = wcol[(size_t)k * NCOLS];       // W[k + 2*half][colN]
        b.y = wcol[(size_t)(k + 1) * NCOLS]; // W[k + 2*half + 1][colN]
        // D = A(16x4,f32) * B(4x16,f32) + C(16x16,f32)
        c = __builtin_amdgcn_wmma_f32_16x16x4_f32(false, a, false, b,
                                                  (short)0, c, false, false);
    }

    const float bv = HAS_BIAS ? bias[colN] : 0.0f;
    float* __restrict__ obase = out + (size_t)(rt * 16 + 8 * half) * NCOLS + colN;
#pragma unroll
    for (int i = 0; i < 8; ++i) {
        // C/D layout: VGPR i holds row (i + 8*half), column lm
        obase[(size_t)i * NCOLS] = c[i] + bv;
    }
}

// ---------------------------------------------------------------------------
// COO SpMM scatter: accum[row[e]] += val[e] * dense[col[e]]
// One wave32 per edge; each lane handles a float4 (32*16B = 512B = one row).
// Hardware f32 atomics at device scope (lowers to global_atomic_add_f32);
// dense operand (20.5 MB) is L2-resident, so gathers are L2 hits.
// ---------------------------------------------------------------------------
__global__ void spmm_scatter_kernel(const int* __restrict__ erow,
                                    const int* __restrict__ ecol,
                                    const float* __restrict__ eval,
                                    const float* __restrict__ dense,
                                    float* __restrict__ accum,
                                    int nedges)
{
    const int lane = threadIdx.x & 31;
    const int e = blockIdx.x * (blockDim.x >> 5) + (threadIdx.x >> 5);
    if (e >= nedges) return;

    const int r = erow[e];
    const int cidx = ecol[e];
    const float v = eval[e];

    const float4 d = ((const float4*)(dense + (size_t)cidx * DIM))[lane];
    float* dst = accum + (size_t)r * DIM + lane * 4;

    __hip_atomic_fetch_add(dst + 0, d.x * v, __ATOMIC_RELAXED, __HIP_MEMORY_SCOPE_AGENT);
    __hip_atomic_fetch_add(dst + 1, d.y * v, __ATOMIC_RELAXED, __HIP_MEMORY_SCOPE_AGENT);
    __hip_atomic_fetch_add(dst + 2, d.z * v, __ATOMIC_RELAXED, __HIP_MEMORY_SCOPE_AGENT);
    __hip_atomic_fetch_add(dst + 3, d.w * v, __ATOMIC_RELAXED, __HIP_MEMORY_SCOPE_AGENT);
}

// accum[n*DIM + d] = bias[d]  (pre-seed SpMM accumulator with the bias)
__global__ void fill_bias_kernel(float* __restrict__ out,
                                 const float* __restrict__ bias,
                                 int total)
{
    const int i = blockIdx.x * blockDim.x + threadIdx.x;
    if (i < total) out[i] = bias[i & (DIM - 1)];
}

__global__ void relu_kernel(float* __restrict__ x, int total)
{
    const int i = blockIdx.x * blockDim.x + threadIdx.x;
    if (i < total) x[i] = fmaxf(x[i], 0.0f);
}

// x = relu(x) + h   (second layer + residual)
__global__ void relu_add_kernel(float* __restrict__ x,
                                const float* __restrict__ h,
                                int total)
{
    const int i = blockIdx.x * blockDim.x + threadIdx.x;
    if (i < total) x[i] = fmaxf(x[i], 0.0f) + h[i];
}

// ---------------------------------------------------------------------------
extern "C" void kernel_launch(void* const* d_in, const int* in_sizes, int n_in,
                              void* d_out, int out_size, void* d_ws, size_t ws_size,
                              hipStream_t stream)
{
    const float* x    = (const float*)d_in[0];
    const int*   erow = (const int*)  d_in[1];
    const int*   ecol = (const int*)  d_in[2];
    const float* ev   = (const float*)d_in[3];
    const float* W1   = (const float*)d_in[4];
    const float* b1   = (const float*)d_in[5];
    const float* W2   = (const float*)d_in[6];
    const float* b2   = (const float*)d_in[7];
    const float* Wf   = (const float*)d_in[8];
    const float* bf   = (const float*)d_in[9];
    float* out = (float*)d_out;

    const int E = in_sizes[1];
    const int total = N_NODES * DIM;
    const size_t SZ = (size_t)N_NODES * DIM;   // elements per [N,128] buffer

    float* bufA = (float*)d_ws;     // GEMM temp (x@W1, then h@W2)
    float* bufB = bufA + SZ;        // layer-1 accumulator -> h
    float* bufC = bufB + SZ;        // layer-2 accumulator -> h2

    const dim3 gemm_grid(N_NODES / 16);               // 2500 row tiles
    const dim3 ew_grid((total + 255) / 256);
    const dim3 spmm_grid((E + 7) / 8);                // 8 edges (waves) per block

    // ---- layer 1: h = relu(adj @ (x @ W1) + b1) ----
    gemm_wmma_f32_kernel<DIM, false><<<gemm_grid, dim3(8 * 32), 0, stream>>>(x, W1, nullptr, bufA);
    fill_bias_kernel<<<ew_grid, 256, 0, stream>>>(bufB, b1, total);
    spmm_scatter_kernel<<<spmm_grid, 256, 0, stream>>>(erow, ecol, ev, bufA, bufB, E);
    relu_kernel<<<ew_grid, 256, 0, stream>>>(bufB, total);

    // ---- layer 2: h2 = relu(adj @ (h @ W2) + b2) + h ----
    gemm_wmma_f32_kernel<DIM, false><<<gemm_grid, dim3(8 * 32), 0, stream>>>(bufB, W2, nullptr, bufA);
    fill_bias_kernel<<<ew_grid, 256, 0, stream>>>(bufC, b2, total);
    spmm_scatter_kernel<<<spmm_grid, 256, 0, stream>>>(erow, ecol, ev, bufA, bufC, E);
    relu_add_kernel<<<ew_grid, 256, 0, stream>>>(bufC, bufB, total);

    // ---- classifier: out = h2 @ Wf + bf  ([N,128]@[128,64]) ----
    gemm_wmma_f32_kernel<64, true><<<gemm_grid, dim3(4 * 32), 0, stream>>>(bufC, Wf, bf, out);
}